// Renderer_87917980549209
// MI455X (gfx1250) — compile-verified
//
#include <hip/hip_runtime.h>
#include <stdint.h>

// Renderer attribute interpolation for MI455X (gfx1250).
// Memory-bound gather+blend: ~155 MB HBM traffic -> ~6.7 us floor @ 23.3 TB/s.
// WMMA is inapplicable (batched 1x3 * 3x32 matvec with per-pixel operands on
// BOTH sides -> no shared contraction operand for 16x16xK WMMA). CDNA5 paths
// used instead: async global->LDS staging (ASYNCcnt + s_wait_asynccnt),
// global_prefetch_b8, non-temporal b128 stores so the 6.4 MB attribute table
// stays resident in the 192 MB L2.

#define TILE_PIX 64   // pixels per block (256 threads = 8 waves)
#define ATTR_D   32   // attribute dim

typedef float v4f __attribute__((ext_vector_type(4)));

__device__ __forceinline__ unsigned lds_off(const void* p) {
  // LDS aperture occupies the high 32 bits of a flat shared pointer;
  // the low 32 bits are the byte offset into LDS.
  return (unsigned)(uintptr_t)p;
}

__device__ __forceinline__ void async_load_b32_to_lds(unsigned lds_byte_off,
                                                      const void* sgpr_base,
                                                      unsigned vgpr_byte_off) {
  // GVS mode: reads the 64-bit base in saddr plus the 32-bit offset in vaddr into the LDS address in vdst
  asm volatile("global_load_async_to_lds_b32 %0, %1, %2"
               :
               : "v"(lds_byte_off), "v"(vgpr_byte_off), "s"(sgpr_base)
               : "memory");
}

__global__ __launch_bounds__(256, 4)
void Renderer_interp_kernel(const int*   __restrict__ pix_to_face,  // (nPix)
                            const float* __restrict__ bary,         // (nPix,3)
                            const int*   __restrict__ faces,        // (F,3)
                            const float* __restrict__ attributes,   // (V,32)
                            float*       __restrict__ out_map,      // (nPix,32)
                            float*       __restrict__ out_mask,     // (nPix)
                            int F, int nPix) {
  __shared__ int   sh_pix[TILE_PIX];
  __shared__ float sh_bary[TILE_PIX * 3];

  const int t       = threadIdx.x;               // 0..255
  const int basePix = blockIdx.x * TILE_PIX;

  // ---- Stage tile inputs into LDS: one async b32 load per lane ----
  if (t < TILE_PIX) {
    const int gp = basePix + t;
    if (gp < nPix) {
      async_load_b32_to_lds(lds_off(&sh_pix[t]), pix_to_face,
                            (unsigned)gp * 4u);
    }
  } else {
    const int j  = t - TILE_PIX;                 // 0..191
    const int gj = basePix * 3 + j;
    if (gj < nPix * 3) {
      async_load_b32_to_lds(lds_off(&sh_bary[j]), bary,
                            (unsigned)gj * 4u);
    }
  }

  // Prefetch next tile's streaming inputs while the async DMA runs.
  if (t == 0) {
    const int np = basePix + TILE_PIX;
    if (np < nPix) {
      __builtin_prefetch(pix_to_face + np, 0, 0);        // global_prefetch_b8
      __builtin_prefetch(bary + (size_t)np * 3, 0, 0);
    }
  }

  asm volatile("s_wait_asynccnt 0" ::: "memory");
  __syncthreads();

  // ---- Compute: wave handles 4 pixels/iter; lane = (pixel-sub, dim-quad) ----
  // lane = sub*8 + quad: sub (0..3) selects pixel in group, quad (0..7)
  // selects a float4 of the 32-dim row -> b128 gathers & stores, fully
  // coalesced (512 B per wave-instruction).
  const int lane = t & 31;
  const int wave = t >> 5;                        // 0..7
  const int quad = lane & 7;                      // dim quad: dims quad*4..+3
  const int sub  = lane >> 3;                     // pixel within group of 4

#pragma unroll
  for (int i = 0; i < TILE_PIX / 32; ++i) {       // 2 iterations
    const int pl = i * 32 + wave * 4 + sub;       // pixel within tile
    const int p  = basePix + pl;
    if (p >= nPix) break;

    const int f = sh_pix[pl];                     // LDS broadcast (8 lanes/addr)
    // jnp.take(..., mode='wrap'): only -1 occurs for background -> F-1
    const unsigned fw = (unsigned)((f < 0) ? (f + F) : f);

    const float b0 = sh_bary[pl * 3 + 0];
    const float b1 = sh_bary[pl * 3 + 1];
    const float b2 = sh_bary[pl * 3 + 2];

    const int* frow = faces + 3ull * fw;
    const int v0 = frow[0];
    const int v1 = frow[1];
    const int v2 = frow[2];

    // b128 row-chunk gathers; attribute table stays resident in L2.
    const v4f a0 = *(const v4f*)(attributes + (uint64_t)v0 * ATTR_D + quad * 4);
    const v4f a1 = *(const v4f*)(attributes + (uint64_t)v1 * ATTR_D + quad * 4);
    const v4f a2 = *(const v4f*)(attributes + (uint64_t)v2 * ATTR_D + quad * 4);

    v4f r;
    r.x = fmaf(b0, a0.x, fmaf(b1, a1.x, b2 * a2.x));
    r.y = fmaf(b0, a0.y, fmaf(b1, a1.y, b2 * a2.y));
    r.z = fmaf(b0, a0.z, fmaf(b1, a1.z, b2 * a2.z));
    r.w = fmaf(b0, a0.w, fmaf(b1, a1.w, b2 * a2.w));

    // Non-temporal b128: 128 MB output must not evict the attribute table.
    __builtin_nontemporal_store(
        r, (v4f*)(out_map + (uint64_t)p * ATTR_D + quad * 4));
  }

  // ---- Mask: coalesced 256 B NT store per block from staged face ids ----
  if (t < TILE_PIX) {
    const int gp = basePix + t;
    if (gp < nPix) {
      __builtin_nontemporal_store((sh_pix[t] != -1) ? 1.0f : 0.0f,
                                  &out_mask[gp]);
    }
  }
}

extern "C" void kernel_launch(void* const* d_in, const int* in_sizes, int n_in,
                              void* d_out, int out_size, void* d_ws, size_t ws_size,
                              hipStream_t stream) {
  const int*   pix_to_face = (const int*)d_in[0];   // (H*W)
  const float* bary        = (const float*)d_in[1]; // (H*W*3)
  const int*   faces       = (const int*)d_in[2];   // (F*3)
  const float* attributes  = (const float*)d_in[3]; // (V*32)

  const int nPix = in_sizes[0];
  const int F    = in_sizes[2] / 3;

  float* out_map  = (float*)d_out;                       // nPix*32 floats
  float* out_mask = out_map + (size_t)nPix * ATTR_D;     // nPix floats

  const int blocks = (nPix + TILE_PIX - 1) / TILE_PIX;
  Renderer_interp_kernel<<<blocks, 256, 0, stream>>>(
      pix_to_face, bary, faces, attributes, out_map, out_mask, F, nPix);
}